// E8Quantizer_59785944760424
// MI455X (gfx1250) — compile-verified
//
#include <hip/hip_runtime.h>

// E8 lattice quantizer for MI455X (gfx1250, wave32).
// HBM-bound stream: 512 MB traffic -> ~22 us floor at 23.3 TB/s. VALU is
// co-critical, so the math is algebraically reduced (parity + distance
// identities), depth-balanced for VOPD dual-issue. Hot loop is a guard-free
// 4-row tile (8 b128 NT loads in flight per thread); a tiny single-row
// remainder loop handles the tail. No matmul structure -> WMMA inapplicable.

typedef __attribute__((ext_vector_type(4))) float v4f;

struct D8Stat {
    float f[8];   // rounded point (pre-nudge)
    float s;      // squared distance of chosen D8 point to input
    float fixp;   // nudge at coord k (0 when no nudge applied)
    int   k;      // argmax coordinate
};

__device__ __forceinline__ void d8_stats(const float x[8], D8Stat& q) {
    float d[8];
#pragma unroll
    for (int i = 0; i < 8; ++i) {
        q.f[i] = rintf(x[i]);          // v_rndne_f32 == jnp.round (half-to-even)
        d[i]   = x[i] - q.f[i];
    }
    // Tournament argmax of |d| (depth 3, earliest-max-wins at every merge,
    // matching jnp.argmax first-occurrence semantics).
    float B[4], DK[4];
    int   K[4];
#pragma unroll
    for (int j = 0; j < 4; ++j) {
        const int i0 = 2 * j, i1 = 2 * j + 1;
        const bool gt = fabsf(d[i1]) > fabsf(d[i0]);
        B[j]  = gt ? fabsf(d[i1]) : fabsf(d[i0]);
        DK[j] = gt ? d[i1] : d[i0];
        K[j]  = gt ? i1 : i0;
    }
    const bool  gA    = B[1] > B[0];
    const float bestA = gA ? B[1]  : B[0];
    const float dkA   = gA ? DK[1] : DK[0];
    const int   kA    = gA ? K[1]  : K[0];
    const bool  gB    = B[3] > B[2];
    const float bestB = gB ? B[3]  : B[2];
    const float dkB   = gB ? DK[3] : DK[2];
    const int   kB    = gB ? K[3]  : K[2];
    const bool  gF    = bestB > bestA;
    const float best  = gF ? bestB : bestA;
    const float dk    = gF ? dkB   : dkA;
    q.k               = gF ? kB    : kA;

    // sum(g) = sum(f) +/- 1  =>  "g even" <=> "sum(f) odd": nudge only if odd.
    // Balanced add tree (exact: integer-valued floats).
    const float sf = ((q.f[0] + q.f[1]) + (q.f[2] + q.f[3])) +
                     ((q.f[4] + q.f[5]) + (q.f[6] + q.f[7]));
    const bool  odd = (((int)sf) & 1) != 0;
    const float fix = (dk < 0.0f) ? -1.0f : 1.0f;
    q.fixp = odd ? fix : 0.0f;

    // Squared distance: sum(d^2); nudged k-th term becomes
    // (d_k - fix)^2 = d_k^2 + 1 - 2|d_k|, with |d_k| == best.
    float s0 = 0.0f, s1 = 0.0f;
#pragma unroll
    for (int i = 0; i < 8; i += 2) {
        s0 = fmaf(d[i],     d[i],     s0);
        s1 = fmaf(d[i + 1], d[i + 1], s1);
    }
    float s = s0 + s1;
    s  += odd ? fmaf(-2.0f, best, 1.0f) : 0.0f;
    q.s = s;
}

__device__ __forceinline__ void e8row(const float x[8], float out[8]) {
    D8Stat q1;
    d8_stats(x, q1);
    float xm[8];
#pragma unroll
    for (int i = 0; i < 8; ++i) xm[i] = x[i] - 0.5f;
    D8Stat q2;
    d8_stats(xm, q2);                 // residual of xm == residual of x vs y2

    const bool  use2 = q2.s < q1.s;   // strict <: keep coset-1 on ties (argmin)
    const float bias = use2 ? 0.5f : 0.0f;          // de-shift folds in here
    const float hit  = (use2 ? q2.fixp : q1.fixp) + bias;
    const int   k    = use2 ? q2.k   : q1.k;
#pragma unroll
    for (int i = 0; i < 8; ++i) {
        const float fi = use2 ? q2.f[i] : q1.f[i];
        out[i] = fi + ((i == k) ? hit : bias);
    }
}

__device__ __forceinline__ void row_compute_store(v4f* __restrict__ out4,
                                                  unsigned int r,
                                                  const v4f a0, const v4f a1) {
    float x[8];
#pragma unroll
    for (int i = 0; i < 4; ++i) { x[i] = a0[i]; x[4 + i] = a1[i]; }
    float y[8];
    e8row(x, y);
    v4f o0, o1;
#pragma unroll
    for (int i = 0; i < 4; ++i) { o0[i] = y[i]; o1[i] = y[4 + i]; }
    __builtin_nontemporal_store(o0, out4 + (size_t)2 * r);
    __builtin_nontemporal_store(o1, out4 + (size_t)2 * r + 1);
}

#define ROWS_PER_ITER 4

__global__ __launch_bounds__(256) void e8_quant_kernel(const float* __restrict__ in,
                                                       float* __restrict__ out,
                                                       unsigned int nrows) {
    const unsigned int tid = blockIdx.x * blockDim.x + threadIdx.x;
    const unsigned int S   = gridDim.x * blockDim.x;
    const v4f* __restrict__ in4  = (const v4f*)in;
    v4f* __restrict__       out4 = (v4f*)out;

    unsigned int r0 = tid;

    // ---- Main loop: full 4-row tiles, no guards, no register init ----
    // (All index values are < ~270M, far from u32 overflow.)
    for (; r0 + 3u * S < nrows; r0 += ROWS_PER_ITER * S) {
        // Prefetch next tile (global_prefetch_b8 on gfx1250).
        const unsigned int rn = r0 + ROWS_PER_ITER * S;
        if (rn < nrows) __builtin_prefetch((const void*)(in4 + (size_t)2 * rn), 0, 0);

        unsigned int r[ROWS_PER_ITER];
        v4f          a[ROWS_PER_ITER][2];
#pragma unroll
        for (int j = 0; j < ROWS_PER_ITER; ++j) {
            r[j]    = r0 + (unsigned int)j * S;
            a[j][0] = __builtin_nontemporal_load(in4 + (size_t)2 * r[j]);
            a[j][1] = __builtin_nontemporal_load(in4 + (size_t)2 * r[j] + 1);
        }
#pragma unroll
        for (int j = 0; j < ROWS_PER_ITER; ++j)
            row_compute_store(out4, r[j], a[j][0], a[j][1]);
    }

    // ---- Remainder: up to 3 single rows per thread (cold, small) ----
    for (; r0 < nrows; r0 += S) {
        const v4f a0 = __builtin_nontemporal_load(in4 + (size_t)2 * r0);
        const v4f a1 = __builtin_nontemporal_load(in4 + (size_t)2 * r0 + 1);
        row_compute_store(out4, r0, a0, a1);
    }
}

extern "C" void kernel_launch(void* const* d_in, const int* in_sizes, int n_in,
                              void* d_out, int out_size, void* d_ws, size_t ws_size,
                              hipStream_t stream) {
    (void)n_in; (void)out_size; (void)d_ws; (void)ws_size;
    const float* in  = (const float*)d_in[0];
    float*       out = (float*)d_out;
    const unsigned int nrows = (unsigned int)(in_sizes[0] / 8);  // 8,388,608 rows

    const int threads = 256;                       // 8 wave32 waves per block
    unsigned long long per_iter = (unsigned long long)threads * ROWS_PER_ITER;
    unsigned long long blk = ((unsigned long long)nrows + per_iter - 1) / per_iter;
    if (blk > 4096) blk = 4096;   // 2 main-loop iterations at this N
    if (blk < 1)    blk = 1;

    e8_quant_kernel<<<(int)blk, threads, 0, stream>>>(in, out, nrows);
}